// FWIForward_54477365182947
// MI455X (gfx1250) — compile-verified
//
#include <hip/hip_runtime.h>

// ---------------- problem constants (fixed by the reference) ----------------
#define NZ    130          // 70 + 2*NBC
#define NX    160          // 100 + 2*NBC
#define NCELL (NZ * NX)    // 20800
#define NGX   40           // float4 groups per row (160/4)
#define NG4   (NZ * NGX)   // 5200 float4 groups
#define NBCI  30
#define NTSTEPS 1000
#define NSHOT 10
#define NG    100
#define NTH   1024
#define PER4  6            // ceil(5200/1024)
#define ISZ   31           // source depth index
#define IGZ   31           // receiver depth index
#define DT_F  0.0008f
#define DX_F  10.0f
#define VNZ   70
#define VNX   100

typedef __attribute__((ext_vector_type(4))) float fl4;

// Ricker wavelet: nw = 2.2/25/0.0008 = 110 -> 111 taps, nc = 55
__device__ __forceinline__ float ricker_at(int t) {
  if (t >= 111) return 0.0f;
  float a = (float)(55 - t) * (25.0f * 0.0008f) * 3.14159265358979f;
  float b = a * a;
  return (1.0f - 2.0f * b) * __expf(-b);
}

// ---------------- main: one persistent workgroup per shot --------------------
__global__ __launch_bounds__(NTH)
void fwi_forward(const float* __restrict__ gAlpha,
                 const float* __restrict__ gK,
                 float* __restrict__ out) {     // [NSHOT][NTSTEPS][NG]
  __shared__ fl4 sPA[NG4];    // ping-pong field A
  __shared__ fl4 sPB[NG4];    // ping-pong field B
  __shared__ fl4 sAl[NG4];    // (v*DT/DX)^2

  const int tid  = threadIdx.x;
  const int shot = blockIdx.x;
  const int isx  = NBCI + 11 * shot;          // source x for this shot
  const int gsrc = ISZ * NGX + (isx >> 2);    // float4 group holding the source
  const int ksrc = isx & 3;

  // ---- stage alpha into LDS (CDNA5 async global->LDS path when available) ----
#if defined(__gfx1250__) && __has_builtin(__builtin_amdgcn_global_load_async_to_lds_b32) && __has_builtin(__builtin_amdgcn_s_wait_asynccnt)
  for (int c = tid; c < NCELL; c += NTH) {
    __builtin_amdgcn_global_load_async_to_lds_b32(
        (__attribute__((address_space(1))) int*)(void*)(gAlpha + c),
        (__attribute__((address_space(3))) int*)(void*)(((float*)sAl) + c),
        0, 0);
  }
  __builtin_amdgcn_s_wait_asynccnt(0);
#else
  for (int c = tid; c < NCELL; c += NTH) ((float*)sAl)[c] = gAlpha[c];
#endif
  const float K = gK[0];    // kappa = K * idx^2

  // zero initial wavefields (vectorized)
  for (int g = tid; g < NG4; g += NTH) { sPA[g] = (fl4)0.0f; sPB[g] = (fl4)0.0f; }
  __syncthreads();

  // src_amp = beta_dt at source = alpha * DX^2
  const float srcAmp = ((float*)sAl)[ISZ * NX + isx] * (DX_F * DX_F);

  fl4* P0 = sPA;   // p0
  fl4* P1 = sPB;   // p1

  const float C2f =  4.0f / 3.0f;
  const float C3f = -1.0f / 12.0f;

  for (int t = 0; t < NTSTEPS; ++t) {
    const float s = ricker_at(t);

    for (int i = 0; i < PER4; ++i) {
      const int g = tid + i * NTH;
      if (g < NG4) {
        const int z  = g / NGX;
        const int gx = g - z * NGX;
        // periodic wrap (matches jnp.roll); group-aligned in x since 160 % 4 == 0
        int zm1 = z - 1; if (zm1 < 0)   zm1 += NZ;
        int zp1 = z + 1; if (zp1 >= NZ) zp1 -= NZ;
        int zm2 = z - 2; if (zm2 < 0)   zm2 += NZ;
        int zp2 = z + 2; if (zp2 >= NZ) zp2 -= NZ;
        int gxm1 = gx - 1; if (gxm1 < 0)    gxm1 += NGX;
        int gxp1 = gx + 1; if (gxp1 >= NGX) gxp1 -= NGX;

        // center row 12-float sliding window (covers x-4 .. x+7 of this group)
        const fl4 cm = P1[z * NGX + gxm1];
        const fl4 cc = P1[z * NGX + gx];
        const fl4 cp = P1[z * NGX + gxp1];
        // z-neighbor rows (aligned b128 loads)
        const fl4 rzm1 = P1[zm1 * NGX + gx];
        const fl4 rzp1 = P1[zp1 * NGX + gx];
        const fl4 rzm2 = P1[zm2 * NGX + gx];
        const fl4 rzp2 = P1[zp2 * NGX + gx];
        const fl4 p0v  = P0[g];
        const fl4 alv  = sAl[g];

        float w[12];
        w[0] = cm[0]; w[1] = cm[1]; w[2]  = cm[2]; w[3]  = cm[3];
        w[4] = cc[0]; w[5] = cc[1]; w[6]  = cc[2]; w[7]  = cc[3];
        w[8] = cp[0]; w[9] = cp[1]; w[10] = cp[2]; w[11] = cp[3];

        // z-part of damping index (x edges take precedence, applied per cell)
        int izq = 0;
        if (z < NBCI)      izq = NBCI - 1 - z;
        if (z >= NZ - NBCI) izq = z - (NZ - NBCI);

        fl4 pn;
#pragma unroll
        for (int k = 0; k < 4; ++k) {
          const int x = 4 * gx + k;
          int idx = izq;
          if (x < NBCI)       idx = NBCI - 1 - x;
          if (x >= NX - NBCI) idx = x - (NX - NBCI);
          const float kap = K * (float)(idx * idx);

          const float lap =
              C2f * (rzm1[k] + rzp1[k] + w[3 + k] + w[5 + k]) +
              C3f * (rzm2[k] + rzp2[k] + w[2 + k] + w[6 + k]);

          const float temp2 = 1.0f - kap;
          const float temp1 = 2.0f - 5.0f * alv[k] - kap;  // 2 + 2*C1*a - kap, C1=-2.5

          float v = temp1 * w[4 + k] - temp2 * p0v[k] + alv[k] * lap;
          if (g == gsrc && k == ksrc) v += srcAmp * s;     // source injection
          pn[k] = v;
        }
        P0[g] = pn;   // overwrite p0 slot: only this thread touches group g of p0
      }
    }
    __syncthreads();   // new field visible; all old-p1 reads complete

    // record receivers from the NEW field
    if (tid < NG) {
      out[(shot * NTSTEPS + t) * NG + tid] = ((float*)P0)[IGZ * NX + (NBCI + tid)];
    }

    // swap: new field becomes p1, old p1 becomes p0 (overwritten next step)
    fl4* tmp = P0; P0 = P1; P1 = tmp;
  }
}

// ---------------- precompute: velmin reduction, alpha field, kappa scale -----
__global__ __launch_bounds__(NTH)
void fwi_precompute(const float* __restrict__ v,
                    float* __restrict__ gAlpha,   // [NCELL]
                    float* __restrict__ gK) {     // [1] kappa scale
  __shared__ float red[NTH];
  const int tid = threadIdx.x;

  // min over raw v (edge padding cannot change the min)
  float m = 1e30f;
  for (int c = tid; c < VNZ * VNX; c += NTH) m = fminf(m, v[c]);
  red[tid] = m;
  __syncthreads();
  for (int off = NTH / 2; off > 0; off >>= 1) {
    if (tid < off) red[tid] = fminf(red[tid], red[tid + off]);
    __syncthreads();
  }
  const float velmin = red[0] * 1000.0f + 3000.0f;

  // alpha = (v_pad * DT / DX)^2 on the edge-replicated padded grid
  for (int c = tid; c < NCELL; c += NTH) {
    int z = c / NX;
    int x = c - z * NX;
    int vz = min(max(z - NBCI, 0), VNZ - 1);
    int vx = min(max(x - NBCI, 0), VNX - 1);
    float vv = v[vz * VNX + vx] * 1000.0f + 3000.0f;
    float a = vv * (DT_F / DX_F);
    gAlpha[c] = a * a;
  }

  // kappa(idx) = k0 * (idx*DX/a)^2 * DT = K * idx^2
  if (tid == 0) {
    float aext = (float)(NBCI - 1) * DX_F;                        // 290
    float k0 = 3.0f * velmin * 16.11809565095832f / (2.0f * aext); // ln(1e7)
    gK[0] = k0 * DT_F * (DX_F / aext) * (DX_F / aext);
  }
}

// ---------------- launch ----------------
extern "C" void kernel_launch(void* const* d_in, const int* in_sizes, int n_in,
                              void* d_out, int out_size, void* d_ws, size_t ws_size,
                              hipStream_t stream) {
  const float* v = (const float*)d_in[0];   // (1,1,70,100) f32
  float* gAlpha = (float*)d_ws;             // NCELL floats
  float* gK     = gAlpha + NCELL;           // 1 float
  float* out    = (float*)d_out;            // (1,10,1000,100) f32

  hipLaunchKernelGGL(fwi_precompute, dim3(1), dim3(NTH), 0, stream, v, gAlpha, gK);
  hipLaunchKernelGGL(fwi_forward, dim3(NSHOT), dim3(NTH), 0, stream, gAlpha, gK, out);
}